// SVD_72189810311961
// MI455X (gfx1250) — compile-verified
//
#include <hip/hip_runtime.h>
#include <hip/hip_bf16.h>
#include <math.h>

#define BATCH   4096
#define NPTS    1024
#define WAVES   4              // waves per block
#define BPW     4              // batches per wave (4x 4x4 moment blocks in one 16x16 WMMA tile)
#define CH      64             // points staged per chunk
#define ROWS    (BPW * 7)      // per batch: w, sx, sy, sz, tx, ty, tz
#define RSTR    68             // LDS row stride in floats (pad: 16B-aligned, spreads banks)
#define NCHUNK  (NPTS / CH)

typedef float v2f __attribute__((ext_vector_type(2)));
typedef float v8f __attribute__((ext_vector_type(8)));
typedef int   v4i_t __attribute__((vector_size(4 * sizeof(int))));
typedef __attribute__((address_space(1))) v4i_t* gptr128_t;  // global  B128
typedef __attribute__((address_space(3))) v4i_t* lptr128_t;  // LDS     B128

#if __has_builtin(__builtin_amdgcn_s_wait_asynccnt)
#define WAIT_ASYNC(n) do { __builtin_amdgcn_s_wait_asynccnt(n); asm volatile("" ::: "memory"); } while (0)
#else
#define WAIT_ASYNC(n) asm volatile("s_wait_asynccnt %0" :: "n"(n) : "memory")
#endif

#if __has_builtin(__builtin_amdgcn_s_wait_dscnt)
#define WAIT_DS(n) do { __builtin_amdgcn_s_wait_dscnt(n); asm volatile("" ::: "memory"); } while (0)
#else
#define WAIT_DS(n) asm volatile("s_wait_dscnt %0" :: "n"(n) : "memory")
#endif

// CDNA5 async global->LDS B128 copy (ASYNCcnt-tracked)
__device__ __forceinline__ void async_b128(const float* g, float* l) {
#if __has_builtin(__builtin_amdgcn_global_load_async_to_lds_b128)
  __builtin_amdgcn_global_load_async_to_lds_b128(
      (gptr128_t)g, (lptr128_t)l, 0, 0);
#else
  unsigned loff = (unsigned)(unsigned long long)(void*)l;
  asm volatile("global_load_async_to_lds_b128 %0, %1, off"
               :: "v"(loff), "v"(g) : "memory");
#endif
}

__global__ __launch_bounds__(WAVES * 32)
void procrustes_wmma_kernel(const float* __restrict__ key,   // (2B,3,N)
                            const float* __restrict__ tgt,   // (B,3,N)
                            const float* __restrict__ wgt,   // (B,1,N)
                            float* __restrict__ out)         // B*9 R ++ B*3 t
{
  __shared__ float sm[2][WAVES][ROWS * RSTR];   // double-buffered stage (60928 B)
  __shared__ float ctile[WAVES][16 * 17];       // C-tile spill (4352 B)
  __shared__ float onesrow[RSTR];               // constant-1 row: kills divergence

  const int tid  = threadIdx.x;
  const int wave = tid >> 5;
  const int lane = tid & 31;
  const int h    = lane >> 4;     // wave half (selects K pair per ISA layout)
  const int r    = lane & 15;     // A-row / B-col index
  const int bbL  = r >> 2;        // sub-batch 0..3
  const int cmp  = r & 3;         // component: 0..2 = x/y/z, 3 = w / ones row

  if (tid < RSTR) onesrow[tid] = 1.0f;

  const int batchBase = (blockIdx.x * WAVES + wave) * BPW;

  // Stage one CH-point chunk for 4 batches: 28 rows x 64 floats = 14 wave-wide B128 issues
  auto issue = [&](int chunk, int buf) {
    const int laneIn = lane & 15;
    const int rh     = lane >> 4;
    const int k0     = chunk * CH;
    float* dst = &sm[buf][wave][0];
#pragma unroll
    for (int q = 0; q < 14; ++q) {
      int rr  = 2 * q + rh;          // staged row 0..27
      int bb  = rr / 7;
      int row = rr % 7;
      int b   = batchBase + bb;
      const float* g;
      if (row == 0)      g = wgt + (size_t)b * NPTS + k0;                               // w
      else if (row <= 3) g = key + ((size_t)(2 * b) * 3 + (row - 1)) * NPTS + k0;       // src xyz
      else               g = tgt + ((size_t)b * 3 + (row - 4)) * NPTS + k0;             // tgt xyz
      async_b128(g + laneIn * 4, dst + rr * RSTR + laneIn * 4);
    }
  };

  v8f acc = {0.f, 0.f, 0.f, 0.f, 0.f, 0.f, 0.f, 0.f};

  issue(0, 0);
  __syncthreads();                 // onesrow visible to all waves
  for (int c = 0; c < NCHUNK; ++c) {
    if (c + 1 < NCHUNK) {
      WAIT_DS(0);                  // prior reads of the buffer we re-fill are drained
      issue(c + 1, (c + 1) & 1);
      WAIT_ASYNC(14);              // chunk c resident; chunk c+1 (14 issues) in flight
    } else {
      WAIT_ASYNC(0);
    }
    const float* S    = &sm[c & 1][wave][0];
    const float* wrow = S + (bbL * 7 + 0) * RSTR;
    // cmp==3 lanes read the constant-1 row: A value = 1*w = w, B value = 1.
    const float* trow = (cmp < 3) ? S + (bbL * 7 + 4 + cmp) * RSTR : onesrow;  // tgt comp (A)
    const float* srow = (cmp < 3) ? S + (bbL * 7 + 1 + cmp) * RSTR : onesrow;  // src comp (B)
#pragma unroll
    for (int kk = 0; kk < CH; kk += 4) {
      const int p0 = kk + 2 * h;   // K pair for this lane half (ISA 16x4 A layout)
      v2f wv = *(const v2f*)(wrow + p0);
      v2f tv = *(const v2f*)(trow + p0);
      v2f sv = *(const v2f*)(srow + p0);
      v2f av = tv * wv;
      // D = A(16x4,f32) x B(4x16,f32) + C(16x16,f32)  — full f32 precision, uniform EXEC
      acc = __builtin_amdgcn_wmma_f32_16x16x4_f32(false, av, false, sv,
                                                  (short)0, acc, false, false);
    }
  }

  // Spill C tile: VGPR v holds row (v + 8*half), col r  (ISA 16x16 f32 C/D layout)
#pragma unroll
  for (int v = 0; v < 8; ++v)
    ctile[wave][(v + 8 * h) * 17 + r] = acc[v];
  __syncthreads();

  // ---- epilogue: 16 threads, one 3x3 SVD + rigid transform each ----
  if (tid < WAVES * BPW) {
    const int w  = tid >> 2;
    const int bb = tid & 3;
    const float* ct = &ctile[w][0];
    float Mom[4][4];
#pragma unroll
    for (int i = 0; i < 4; ++i)
#pragma unroll
      for (int j = 0; j < 4; ++j)
        Mom[i][j] = ct[(4 * bb + i) * 17 + (4 * bb + j)];

    const float wsum = Mom[3][3];
    const float inv  = 1.0f / wsum;
    float tc[3], sc[3], Hm[3][3];
#pragma unroll
    for (int i = 0; i < 3; ++i) { tc[i] = Mom[i][3] * inv; sc[i] = Mom[3][i] * inv; }
#pragma unroll
    for (int i = 0; i < 3; ++i)
#pragma unroll
      for (int j = 0; j < 3; ++j)
        Hm[i][j] = Mom[i][j] * inv - tc[i] * sc[j];

    // Jacobi eigensolve of A = H^T H  ->  V, eigenvalues on diagonal
    float Am[3][3], Vm[3][3];
#pragma unroll
    for (int i = 0; i < 3; ++i)
#pragma unroll
      for (int j = 0; j < 3; ++j) {
        Am[i][j] = Hm[0][i] * Hm[0][j] + Hm[1][i] * Hm[1][j] + Hm[2][i] * Hm[2][j];
        Vm[i][j] = (i == j) ? 1.0f : 0.0f;
      }
    for (int sweep = 0; sweep < 8; ++sweep) {
#pragma unroll
      for (int pq = 0; pq < 3; ++pq) {
        const int p = (pq == 2) ? 1 : 0;
        const int q = (pq == 0) ? 1 : 2;
        float apq = Am[p][q];
        if (fabsf(apq) > 1e-20f) {
          float theta = (Am[q][q] - Am[p][p]) / (2.0f * apq);
          float t = 1.0f / (fabsf(theta) + sqrtf(theta * theta + 1.0f));
          t = (theta >= 0.0f) ? t : -t;
          float cth = 1.0f / sqrtf(t * t + 1.0f);
          float sth = t * cth;
#pragma unroll
          for (int k = 0; k < 3; ++k) {
            float akp = Am[k][p], akq = Am[k][q];
            Am[k][p] = cth * akp - sth * akq;
            Am[k][q] = sth * akp + cth * akq;
          }
#pragma unroll
          for (int k = 0; k < 3; ++k) {
            float apk = Am[p][k], aqk = Am[q][k];
            Am[p][k] = cth * apk - sth * aqk;
            Am[q][k] = sth * apk + cth * aqk;
          }
#pragma unroll
          for (int k = 0; k < 3; ++k) {
            float vkp = Vm[k][p], vkq = Vm[k][q];
            Vm[k][p] = cth * vkp - sth * vkq;
            Vm[k][q] = sth * vkp + cth * vkq;
          }
        }
      }
    }

    // top-2 eigenpairs
    float l0 = Am[0][0], l1 = Am[1][1], l2 = Am[2][2];
    int i0, i1;
    if (l0 >= l1 && l0 >= l2)      { i0 = 0; i1 = (l1 >= l2) ? 1 : 2; }
    else if (l1 >= l0 && l1 >= l2) { i0 = 1; i1 = (l0 >= l2) ? 0 : 2; }
    else                           { i0 = 2; i1 = (l0 >= l1) ? 0 : 1; }
    float v0[3] = {Vm[0][i0], Vm[1][i0], Vm[2][i0]};
    float v1[3] = {Vm[0][i1], Vm[1][i1], Vm[2][i1]};

    float u0[3], u1[3];
#pragma unroll
    for (int i = 0; i < 3; ++i)
      u0[i] = Hm[i][0] * v0[0] + Hm[i][1] * v0[1] + Hm[i][2] * v0[2];
    float n0 = 1.0f / sqrtf(u0[0]*u0[0] + u0[1]*u0[1] + u0[2]*u0[2]);
#pragma unroll
    for (int i = 0; i < 3; ++i) u0[i] *= n0;
#pragma unroll
    for (int i = 0; i < 3; ++i)
      u1[i] = Hm[i][0] * v1[0] + Hm[i][1] * v1[1] + Hm[i][2] * v1[2];
    float d01 = u0[0]*u1[0] + u0[1]*u1[1] + u0[2]*u1[2];
#pragma unroll
    for (int i = 0; i < 3; ++i) u1[i] -= d01 * u0[i];
    float n1 = 1.0f / sqrtf(u1[0]*u1[0] + u1[1]*u1[1] + u1[2]*u1[2]);
#pragma unroll
    for (int i = 0; i < 3; ++i) u1[i] *= n1;

    float u2[3] = {u0[1]*u1[2] - u0[2]*u1[1],
                   u0[2]*u1[0] - u0[0]*u1[2],
                   u0[0]*u1[1] - u0[1]*u1[0]};
    float w2[3] = {v0[1]*v1[2] - v0[2]*v1[1],
                   v0[2]*v1[0] - v0[0]*v1[2],
                   v0[0]*v1[1] - v0[1]*v1[0]};

    // R = u0 v0^T + u1 v1^T + (u0 x u1)(v0 x v1)^T  ==  U diag(1,1,det UV) Vh
    float Rm[3][3];
#pragma unroll
    for (int i = 0; i < 3; ++i)
#pragma unroll
      for (int j = 0; j < 3; ++j)
        Rm[i][j] = u0[i]*v0[j] + u1[i]*v1[j] + u2[i]*w2[j];

    // t1 = src_centroid - R^T tgt_centroid ; t_out = -R t1
    float t1[3], to[3];
#pragma unroll
    for (int i = 0; i < 3; ++i)
      t1[i] = sc[i] - (Rm[0][i]*tc[0] + Rm[1][i]*tc[1] + Rm[2][i]*tc[2]);
#pragma unroll
    for (int i = 0; i < 3; ++i)
      to[i] = -(Rm[i][0]*t1[0] + Rm[i][1]*t1[1] + Rm[i][2]*t1[2]);

    const int b = blockIdx.x * (WAVES * BPW) + tid;
    float* Ro = out + (size_t)b * 9;
    float* To = out + (size_t)BATCH * 9 + (size_t)b * 3;
#pragma unroll
    for (int i = 0; i < 3; ++i) {
#pragma unroll
      for (int j = 0; j < 3; ++j) Ro[i * 3 + j] = Rm[i][j];
      To[i] = to[i];
    }
  }
}

extern "C" void kernel_launch(void* const* d_in, const int* in_sizes, int n_in,
                              void* d_out, int out_size, void* d_ws, size_t ws_size,
                              hipStream_t stream) {
  const float* key = (const float*)d_in[0];   // (2B,3,N)
  const float* tgt = (const float*)d_in[1];   // (B,3,N)
  const float* wgt = (const float*)d_in[2];   // (B,1,N)
  float* out = (float*)d_out;                 // B*9 ++ B*3
  dim3 grid(BATCH / (WAVES * BPW));           // 256 blocks
  dim3 block(WAVES * 32);                     // 128 threads = 4 waves
  hipLaunchKernelGGL(procrustes_wmma_kernel, grid, block, 0, stream,
                     key, tgt, wgt, out);
}